// GraphSAGE_74251394613579
// MI455X (gfx1250) — compile-verified
//
#include <hip/hip_runtime.h>
#include <hip/hip_bf16.h>

#define N_NODES 40000
#define N_EDGES 640000
#define IN_CH   128
#define HID_CH  128
#define OUT_CH  32

typedef float v2f __attribute__((ext_vector_type(2)));
typedef float v8f __attribute__((ext_vector_type(8)));

// ---------------------------------------------------------------------------
// Zero-fill (graph-capture-safe replacement for hipMemsetAsync)
// ---------------------------------------------------------------------------
__global__ __launch_bounds__(256) void zero_f32(float* __restrict__ p, int n) {
    int i = blockIdx.x * 256 + threadIdx.x;
    if (i < n) p[i] = 0.0f;
}

// ---------------------------------------------------------------------------
// Degree: deg[dst[e]] += 1  (fp32 atomics, L2-resident)
// ---------------------------------------------------------------------------
__global__ __launch_bounds__(256) void degree_kernel(const int* __restrict__ dst,
                                                     float* __restrict__ deg) {
    int e = blockIdx.x * 256 + threadIdx.x;
    if (e < N_EDGES) atomicAdd(&deg[dst[e]], 1.0f);
}

// ---------------------------------------------------------------------------
// Scatter-add of 128-channel rows: one wave32 per edge, 4 channels per lane.
// Coalesced float4 read of x[src], 4 global_atomic_add_f32 into agg[dst].
// ---------------------------------------------------------------------------
__global__ __launch_bounds__(256) void scatter_kernel(const float* __restrict__ x,
                                                      const int*   __restrict__ src,
                                                      const int*   __restrict__ dst,
                                                      float*       __restrict__ agg) {
    long t    = (long)blockIdx.x * 256 + threadIdx.x;
    int  e    = (int)(t >> 5);          // wave-uniform edge id
    int  lane = (int)(t & 31);
    if (e >= N_EDGES) return;
    int s = src[e];
    int d = dst[e];
    float4 v = ((const float4*)(x + (long)s * 128))[lane];
    float* ad = agg + (long)d * 128 + lane * 4;
    atomicAdd(ad + 0, v.x);
    atomicAdd(ad + 1, v.y);
    atomicAdd(ad + 2, v.z);
    atomicAdd(ad + 3, v.w);
}

// ---------------------------------------------------------------------------
// Fused SAGE linear layer with LDS-staged A operands:
//   y = [relu]( (agg/max(deg,1)) @ Wl^T + b + x @ Wr^T )
//
// Block = TM M-tiles x TN N-tiles, one wave32 per 16x16 output tile.
// A tiles (mean-scaled agg, and x) are staged to LDS with coalesced float4
// loads (mean division folded in), then read conflict-free in the K loop
// (row stride 132 floats: lanes 0-15 -> banks 4r, lanes 16-31 -> banks 4r+2).
// Two independent accumulators break the serial WMMA dependency chain.
//
// Operand layouts per CDNA5 ISA (V_WMMA_F32_16X16X4_F32):
//   A (16x4 f32, 2 VGPRs):  row = lane&15, K = (lane>>4)*2 + {0,1}
//   B (4x16 f32, 2 VGPRs):  col = lane&15, K = (lane>>4)*2 + {0,1}
//   C/D (16x16 f32, 8 VGPRs): VGPR v -> M = v + (lane>>4)*8, N = lane&15
// ---------------------------------------------------------------------------
__global__ __launch_bounds__(256) void sage_linear_lds(
        const float* __restrict__ agg,   // [N, Cin]
        const float* __restrict__ deg,   // [N]
        const float* __restrict__ x,     // [N, Cin]
        const float* __restrict__ Wl,    // [Cout, Cin]
        const float* __restrict__ bias,  // [Cout]
        const float* __restrict__ Wr,    // [Cout, Cin]
        float*       __restrict__ y,     // [N, Cout]
        int Cin, int Cout, int TM, int TN, int do_relu) {
    extern __shared__ float smem[];
    const int LDS_STRIDE = 132;              // +4 pad -> conflict-free ds reads
    const int TILE_FL    = 16 * LDS_STRIDE;  // floats per staged 16-row tile

    const int tid      = threadIdx.x;
    const int nthreads = blockDim.x;
    const int lane     = tid & 31;
    const int wave     = tid >> 5;

    // ---- stage A tiles into LDS (mean scaling fused), coalesced float4 ----
    const int  f4_per_row = Cin >> 2;
    const int  total_f4   = TM * 16 * f4_per_row;
    const long node_base  = (long)blockIdx.x * TM * 16;
    for (int idx = tid; idx < total_f4; idx += nthreads) {
        int  row  = idx / f4_per_row;        // 0 .. TM*16-1
        int  col4 = idx - row * f4_per_row;
        long node = node_base + row;
        float rdeg = 1.0f / fmaxf(deg[node], 1.0f);
        float4 a = ((const float4*)(agg + node * Cin))[col4];
        float4 b = ((const float4*)(x   + node * Cin))[col4];
        a.x *= rdeg; a.y *= rdeg; a.z *= rdeg; a.w *= rdeg;
        int mloc = row >> 4;
        int r    = row & 15;
        float* dA = smem + mloc * 2 * TILE_FL + r * LDS_STRIDE + col4 * 4;
        *(float4*)dA             = a;        // mean operand
        *(float4*)(dA + TILE_FL) = b;        // root operand
    }
    __syncthreads();

    // ---- per-wave tile assignment (wave-uniform, EXEC all-1s for WMMA) ----
    const int mloc = wave / TN;
    const int nt   = wave - mloc * TN;
    const int r    = lane & 15;
    const int koff = (lane >> 4) * 2;

    const float* As = smem + mloc * 2 * TILE_FL + r * LDS_STRIDE + koff;
    const float* Xs = As + TILE_FL;
    const int rowB  = nt * 16 + r;
    const float* bL = Wl + (long)rowB * Cin + koff;
    const float* bR = Wr + (long)rowB * Cin + koff;

    v8f accL = {}, accR = {};
    for (int k0 = 0; k0 < Cin; k0 += 4) {
        v2f a1 = *(const v2f*)(As + k0);     // ds_load_b64, conflict-free
        v2f b1; b1.x = bL[k0]; b1.y = bL[k0 + 1];
        accL = __builtin_amdgcn_wmma_f32_16x16x4_f32(
            false, a1, false, b1, (short)0, accL, false, false);

        v2f a2 = *(const v2f*)(Xs + k0);
        v2f b2; b2.x = bR[k0]; b2.y = bR[k0 + 1];
        accR = __builtin_amdgcn_wmma_f32_16x16x4_f32(
            false, a2, false, b2, (short)0, accR, false, false);
    }

    // ---- epilogue: merge chains, bias, relu, store ----
    const int   n     = nt * 16 + r;
    const float bn    = bias[n];
    const long  mbase = node_base + mloc * 16 + (lane >> 4) * 8;
    #pragma unroll
    for (int v = 0; v < 8; ++v) {
        float val = accL[v] + accR[v] + bn;
        if (do_relu) val = fmaxf(val, 0.0f);
        y[(mbase + v) * Cout + n] = val;
    }
}

// ---------------------------------------------------------------------------
// log_softmax over 32 channels: one wave32 per node, lane == channel.
// ---------------------------------------------------------------------------
__global__ __launch_bounds__(256) void log_softmax_kernel(float* __restrict__ y) {
    int node = blockIdx.x * 8 + (threadIdx.x >> 5);
    int lane = threadIdx.x & 31;
    if (node >= N_NODES) return;
    float v = y[(long)node * 32 + lane];
    float m = v;
    #pragma unroll
    for (int o = 16; o > 0; o >>= 1) m = fmaxf(m, __shfl_xor(m, o, 32));
    float e = expf(v - m);
    float s = e;
    #pragma unroll
    for (int o = 16; o > 0; o >>= 1) s += __shfl_xor(s, o, 32);
    y[(long)node * 32 + lane] = (v - m) - logf(s);
}

// ---------------------------------------------------------------------------
// Host orchestration
// ---------------------------------------------------------------------------
extern "C" void kernel_launch(void* const* d_in, const int* in_sizes, int n_in,
                              void* d_out, int out_size, void* d_ws, size_t ws_size,
                              hipStream_t stream) {
    const float* x    = (const float*)d_in[0];
    const int*   ei   = (const int*)  d_in[1];   // [2, N_EDGES]
    const float* W1l  = (const float*)d_in[2];
    const float* b1   = (const float*)d_in[3];
    const float* W1r  = (const float*)d_in[4];
    const float* W2l  = (const float*)d_in[5];
    const float* b2   = (const float*)d_in[6];
    const float* W2r  = (const float*)d_in[7];
    const float* W3l  = (const float*)d_in[8];
    const float* b3   = (const float*)d_in[9];
    const float* W3r  = (const float*)d_in[10];
    float*       out  = (float*)d_out;           // [N_NODES, 32]

    const int* src = ei;
    const int* dst = ei + N_EDGES;

    // Workspace layout (floats): deg | agg | h1 | h2
    float* deg = (float*)d_ws;
    float* agg = deg + N_NODES;
    float* h1  = agg + (long)N_NODES * HID_CH;
    float* h2  = h1  + (long)N_NODES * HID_CH;

    const int ROWFL = N_NODES * HID_CH;                  // 5,120,000
    const dim3 blk(256);
    const int  scat_blocks = (N_EDGES * 32 + 255) / 256; // wave per edge

    // LDS bytes: TM * 2 matrices * 16 rows * 132 floats * 4B
    const size_t lds_1 = (size_t)1 * 2 * 16 * 132 * 4;   // 16,896 B
    const size_t lds_3 = (size_t)2 * 2 * 16 * 132 * 4;   // 33,792 B

    // deg + agg are contiguous: zero both in one sweep, compute deg once.
    zero_f32<<<(N_NODES + ROWFL + 255) / 256, blk, 0, stream>>>(deg, N_NODES + ROWFL);
    degree_kernel<<<(N_EDGES + 255) / 256, blk, 0, stream>>>(dst, deg);

    // ---- Layer 1: x -> h1 (128 -> 128, relu). 2500 blocks, 1 Mtile x 8 Ntiles
    scatter_kernel<<<scat_blocks, blk, 0, stream>>>(x, src, dst, agg);
    sage_linear_lds<<<N_NODES / 16, blk, lds_1, stream>>>(
        agg, deg, x, W1l, b1, W1r, h1, IN_CH, HID_CH, 1, 8, 1);

    // ---- Layer 2: h1 -> h2 (128 -> 128, relu) ----
    zero_f32<<<(ROWFL + 255) / 256, blk, 0, stream>>>(agg, ROWFL);
    scatter_kernel<<<scat_blocks, blk, 0, stream>>>(h1, src, dst, agg);
    sage_linear_lds<<<N_NODES / 16, blk, lds_1, stream>>>(
        agg, deg, h1, W2l, b2, W2r, h2, HID_CH, HID_CH, 1, 8, 1);

    // ---- Layer 3: h2 -> out (128 -> 32, no relu). 1250 blocks of 4 waves:
    //      2 Mtiles x 2 Ntiles.
    zero_f32<<<(ROWFL + 255) / 256, blk, 0, stream>>>(agg, ROWFL);
    scatter_kernel<<<scat_blocks, blk, 0, stream>>>(h2, src, dst, agg);
    sage_linear_lds<<<N_NODES / 32, dim3(128), lds_3, stream>>>(
        agg, deg, h2, W3l, b3, W3r, out, HID_CH, OUT_CH, 2, 2, 0);

    // ---- log_softmax over 32 channels ----
    log_softmax_kernel<<<(N_NODES + 7) / 8, blk, 0, stream>>>(out);
}